// MultiTaskDeep4Net_10393820857105
// MI455X (gfx1250) — compile-verified
//
#include <hip/hip_runtime.h>
#include <math.h>

typedef float v2f __attribute__((ext_vector_type(2)));
typedef float v8f __attribute__((ext_vector_type(8)));

#define BEEG   128
#define CCH    64
#define TT     750
#define KPADC  12          // per-channel K padding (10 taps -> 12)
#define KTOT   (CCH*KPADC) // 768
#define T1     741         // conv1 valid length
#define NBLK_T 12          // ceil(741/64) 64-wide time blocks
#define LDS_TSTRIDE 80     // LDS row stride (76 staged + pad)
#define LDS_TSTAGE  76     // staged t values per channel (64 + 11 halo + 1)

// ---------------- setup: fold conv_t*conv_s + bn1 into W1T[K][32] ----------------
__global__ void fold_w1(const float* __restrict__ ct_w, const float* __restrict__ cs_w,
                        const float* __restrict__ g, const float* __restrict__ v,
                        float* __restrict__ w1t) {
    int idx = blockIdx.x * 256 + threadIdx.x;
    if (idx >= KTOT * 32) return;
    int o = idx & 31;
    int kidx = idx >> 5;
    int c = kidx / KPADC, k = kidx - c * KPADC;
    float val = 0.f;
    if (o < 25 && k < 10) {
        float s = g[o] * rsqrtf(v[o] + 1e-5f);
        float acc = 0.f;
        for (int i = 0; i < 25; ++i)
            acc += cs_w[(o * 25 + i) * CCH + c] * ct_w[i * 10 + k];
        val = s * acc;
    }
    w1t[idx] = val;   // K-major: w1t[kidx*32 + o]
}

__global__ void fold_b1(const float* __restrict__ ct_b, const float* __restrict__ cs_w,
                        const float* __restrict__ g, const float* __restrict__ bvec,
                        const float* __restrict__ m, const float* __restrict__ v,
                        float* __restrict__ bias1) {
    int o = threadIdx.x;
    if (o >= 32) return;
    float val = 0.f;
    if (o < 25) {
        float s = g[o] * rsqrtf(v[o] + 1e-5f);
        float acc = 0.f;
        for (int i = 0; i < 25; ++i) {
            float sc = 0.f;
            for (int c = 0; c < CCH; ++c) sc += cs_w[(o * 25 + i) * CCH + c];
            acc += sc * ct_b[i];
        }
        val = s * (acc - m[o]) + bvec[o];
    }
    bias1[o] = val;
}

// ---------------- setup: fold bn into conv weights (layers 2-4) ----------------
__global__ void scale_w(const float* __restrict__ w, const float* __restrict__ g,
                        const float* __restrict__ v, float* __restrict__ out,
                        int OC, int ICK) {
    int idx = blockIdx.x * 256 + threadIdx.x;
    if (idx >= OC * ICK) return;
    int o = idx / ICK;
    out[idx] = w[idx] * g[o] * rsqrtf(v[o] + 1e-5f);
}

__global__ void fold_bias(const float* __restrict__ g, const float* __restrict__ bvec,
                          const float* __restrict__ m, const float* __restrict__ v,
                          float* __restrict__ out, int OC) {
    int o = blockIdx.x * 64 + threadIdx.x;
    if (o >= OC) return;
    out[o] = bvec[o] - m[o] * g[o] * rsqrtf(v[o] + 1e-5f);
}

// ---------------- layer 1: fused conv_t+conv_s+bn1+elu via f32 WMMA ----------------
// Block = 128 threads (4 waves) covering one sample x 64 time columns.
// x tile (64ch x 76t = 20KB) staged to LDS with GLOBAL_LOAD_ASYNC_TO_LDS_B32.
// Each wave owns a 16-wide N tile and computes BOTH 16-row M tiles (M=32 total),
// sharing one B fragment per K-step -> 2 WMMA per 2 ds-loads.
__global__ __launch_bounds__(128) void conv1_wmma(const float* __restrict__ x,
                                                  const float* __restrict__ w1t,
                                                  const float* __restrict__ bias1,
                                                  float* __restrict__ a1full) {
    __shared__ float xs[CCH * LDS_TSTRIDE];   // 20480 floats? no: 64*80=5120 floats = 20KB

    const int tid  = threadIdx.x;
    const int lane = tid & 31;
    const int wave = tid >> 5;
    const int b    = blockIdx.x / NBLK_T;
    const int blk  = blockIdx.x - b * NBLK_T;
    const int t0blk = blk * 64;
    const int n    = lane & 15;
    const int half = lane >> 4;
    const int tloc = wave * 16;

    // ---- async stage x[b][c][t0blk .. t0blk+75] -> xs[c][0..75] (ASYNCcnt path) ----
    {
        const float* xbase = x + b * (CCH * TT);          // uniform SGPR pair
        const unsigned lds_base = (unsigned)(uintptr_t)xs; // LDS byte offset of xs
        // 64*76 = 4864 dwords, 128 threads -> 38 full iterations (no divergence)
        for (int j = tid; j < CCH * LDS_TSTAGE; j += 128) {
            int c  = j / LDS_TSTAGE;
            int jj = j - c * LDS_TSTAGE;
            int t  = t0blk + jj;
            t = t > (TT - 1) ? (TT - 1) : t;              // clamp: only feeds zero-weight/garbage cols
            unsigned gofs = (unsigned)(c * TT + t) * 4u;  // byte offset from xbase (GVS mode)
            unsigned lofs = lds_base + (unsigned)(c * LDS_TSTRIDE + jj) * 4u;
            asm volatile("global_load_async_to_lds_b32 %0, %1, %2"
                         :: "v"(lofs), "v"(gofs), "s"(xbase) : "memory");
        }
        asm volatile("s_wait_asynccnt 0x0" ::: "memory");
    }
    __syncthreads();

    // ---- WMMA main loop: D = W1T^T(32x768) * B(768x16) ----
    v8f acc0 = {};   // rows 0-15
    v8f acc1 = {};   // rows 16-31 (only 16-24 real)
    for (int kb = 0; kb < KTOT; kb += 4) {
        const int ka0 = kb + half * 2;
        const int ka1 = ka0 + 1;
        v2f a0, a1;
        a0.x = w1t[ka0 * 32 + n];            // M rows 0-15  (L0-hot, coalesced)
        a0.y = w1t[ka1 * 32 + n];
        a1.x = w1t[ka0 * 32 + 16 + n];       // M rows 16-31
        a1.y = w1t[ka1 * 32 + 16 + n];
        const int c0 = ka0 / KPADC, k0 = ka0 - c0 * KPADC;
        const int c1 = ka1 / KPADC, k1 = ka1 - c1 * KPADC;
        v2f bm;                               // B fragment from LDS, shared by both tiles
        bm.x = xs[c0 * LDS_TSTRIDE + tloc + n + k0];
        bm.y = xs[c1 * LDS_TSTRIDE + tloc + n + k1];
        acc0 = __builtin_amdgcn_wmma_f32_16x16x4_f32(false, a0, false, bm, (short)0, acc0, false, false);
        acc1 = __builtin_amdgcn_wmma_f32_16x16x4_f32(false, a1, false, bm, (short)0, acc1, false, false);
    }

    // ---- bias + ELU + guarded store ----
    const int col = t0blk + tloc + n;
#pragma unroll
    for (int r = 0; r < 8; ++r) {
        const int m0 = half ? r + 8 : r;      // 0..15, always valid channel
        if (col < T1) {
            float val = acc0[r] + bias1[m0];
            val = val > 0.f ? val : (expf(val) - 1.f);
            a1full[(b * 25 + m0) * T1 + col] = val;
            const int m1 = m0 + 16;           // 16..31, valid only < 25
            if (m1 < 25) {
                float v1 = acc1[r] + bias1[m1];
                v1 = v1 > 0.f ? v1 : (expf(v1) - 1.f);
                a1full[(b * 25 + m1) * T1 + col] = v1;
            }
        }
    }
}

// ---------------- pool1: max over 3, stride 3 ----------------
__global__ void pool1(const float* __restrict__ a1full, float* __restrict__ a1p) {
    int idx = blockIdx.x * 256 + threadIdx.x;
    if (idx >= BEEG * 25 * 247) return;
    int tp = idx % 247;
    int rest = idx / 247;                     // b*25 + m
    const float* p = a1full + rest * T1 + tp * 3;
    a1p[idx] = fmaxf(p[0], fmaxf(p[1], p[2]));
}

// ---------------- generic fused conv(10-tap)+foldedBN+elu+pool3 ----------------
__global__ void conv_pool(const float* __restrict__ in, const float* __restrict__ w,
                          const float* __restrict__ bias, float* __restrict__ out,
                          int IC, int Tin, int OC, int Tpo) {
    int idx = blockIdx.x * 256 + threadIdx.x;
    if (idx >= BEEG * OC * Tpo) return;
    int tp = idx % Tpo;
    int r  = idx / Tpo;
    int o  = r % OC;
    int b  = r / OC;
    const float* inb = in + b * IC * Tin;
    const float* wo  = w + o * IC * 10;
    float best = -INFINITY;
    for (int wnd = 0; wnd < 3; ++wnd) {
        int t = tp * 3 + wnd;
        float acc = bias[o];
        for (int i = 0; i < IC; ++i) {
            const float* ip = inb + i * Tin + t;
            const float* wp = wo + i * 10;
#pragma unroll
            for (int k = 0; k < 10; ++k) acc = fmaf(wp[k], ip[k], acc);
        }
        float e = acc > 0.f ? acc : (expf(acc) - 1.f);
        best = fmaxf(best, e);
    }
    out[idx] = best;
}

// ---------------- fc + per-sample MoE head (one block per sample) ----------------
__global__ __launch_bounds__(128) void fc_heads(const float* __restrict__ a4,
                                                const float* __restrict__ fc_w,
                                                const float* __restrict__ fc_b,
                                                const int* __restrict__ cid,
                                                const float* __restrict__ hW1,
                                                const float* __restrict__ hb1,
                                                const float* __restrict__ hW2,
                                                const float* __restrict__ hb2,
                                                float* __restrict__ out) {
    __shared__ float sf[4];
    __shared__ float sh[128];
    const int b = blockIdx.x;
    const int j = threadIdx.x;
    const int c = cid[b];
    if (j < 4) {
        float acc = fc_b[j];
        const float* ab = a4 + b * 800;
        const float* wp = fc_w + j * 800;
        for (int t = 0; t < 800; ++t) acc = fmaf(wp[t], ab[t], acc);
        sf[j] = acc;
    }
    __syncthreads();
    {
        float acc = hb1[c * 128 + j];
#pragma unroll
        for (int f = 0; f < 4; ++f) acc = fmaf(sf[f], hW1[(c * 4 + f) * 128 + j], acc);
        sh[j] = acc > 0.f ? acc : 0.f;
    }
    __syncthreads();
    if (j < 4) {
        float acc = hb2[c * 4 + j];
        for (int t = 0; t < 128; ++t) acc = fmaf(sh[t], hW2[(c * 128 + t) * 4 + j], acc);
        out[b * 4 + j] = acc;
    }
}

extern "C" void kernel_launch(void* const* d_in, const int* in_sizes, int n_in,
                              void* d_out, int out_size, void* d_ws, size_t ws_size,
                              hipStream_t stream) {
    (void)in_sizes; (void)n_in; (void)out_size; (void)ws_size;
    const float* x     = (const float*)d_in[0];
    const int*   cid   = (const int*)  d_in[1];
    const float* ct_w  = (const float*)d_in[2];
    const float* ct_b  = (const float*)d_in[3];
    const float* cs_w  = (const float*)d_in[4];
    const float* bn1_g = (const float*)d_in[5];
    const float* bn1_b = (const float*)d_in[6];
    const float* bn1_m = (const float*)d_in[7];
    const float* bn1_v = (const float*)d_in[8];
    const float* c2_w  = (const float*)d_in[9];
    const float* bn2_g = (const float*)d_in[10];
    const float* bn2_b = (const float*)d_in[11];
    const float* bn2_m = (const float*)d_in[12];
    const float* bn2_v = (const float*)d_in[13];
    const float* c3_w  = (const float*)d_in[14];
    const float* bn3_g = (const float*)d_in[15];
    const float* bn3_b = (const float*)d_in[16];
    const float* bn3_m = (const float*)d_in[17];
    const float* bn3_v = (const float*)d_in[18];
    const float* c4_w  = (const float*)d_in[19];
    const float* bn4_g = (const float*)d_in[20];
    const float* bn4_b = (const float*)d_in[21];
    const float* bn4_m = (const float*)d_in[22];
    const float* bn4_v = (const float*)d_in[23];
    const float* fc_w  = (const float*)d_in[24];
    const float* fc_b  = (const float*)d_in[25];
    const float* hW1   = (const float*)d_in[26];
    const float* hb1   = (const float*)d_in[27];
    const float* hW2   = (const float*)d_in[28];
    const float* hb2   = (const float*)d_in[29];
    float* out = (float*)d_out;

    // workspace layout (floats), total ~4.35M floats (~17.4 MB)
    float* ws   = (float*)d_ws;
    float* W1T  = ws;                 // 768*32 = 24576
    float* b1   = W1T + 24576;        // 32
    float* a1f  = b1  + 32;           // 128*25*741 = 2371200
    float* a1p  = a1f + 2371200;      // 128*25*247 = 790400
    float* w2f  = a1p + 790400;       // 12500
    float* b2f  = w2f + 12500;        // 50
    float* a2   = b2f + 50;           // 128*50*79 = 505600
    float* w3f  = a2  + 505600;       // 50000
    float* b3f  = w3f + 50000;        // 100
    float* a3   = b3f + 100;          // 128*100*23 = 294400
    float* w4f  = a3  + 294400;       // 200000
    float* b4f  = w4f + 200000;       // 200
    float* a4   = b4f + 200;          // 128*200*4 = 102400

    // --- fold weights ---
    fold_w1<<<(KTOT * 32 + 255) / 256, 256, 0, stream>>>(ct_w, cs_w, bn1_g, bn1_v, W1T);
    fold_b1<<<1, 32, 0, stream>>>(ct_b, cs_w, bn1_g, bn1_b, bn1_m, bn1_v, b1);
    scale_w<<<(12500 + 255) / 256, 256, 0, stream>>>(c2_w, bn2_g, bn2_v, w2f, 50, 250);
    fold_bias<<<1, 64, 0, stream>>>(bn2_g, bn2_b, bn2_m, bn2_v, b2f, 50);
    scale_w<<<(50000 + 255) / 256, 256, 0, stream>>>(c3_w, bn3_g, bn3_v, w3f, 100, 500);
    fold_bias<<<2, 64, 0, stream>>>(bn3_g, bn3_b, bn3_m, bn3_v, b3f, 100);
    scale_w<<<(200000 + 255) / 256, 256, 0, stream>>>(c4_w, bn4_g, bn4_v, w4f, 200, 1000);
    fold_bias<<<4, 64, 0, stream>>>(bn4_g, bn4_b, bn4_m, bn4_v, b4f, 200);

    // --- layer 1: WMMA GEMM with async-LDS staging (1536 blocks x 4 waves) ---
    conv1_wmma<<<BEEG * NBLK_T, 128, 0, stream>>>(x, W1T, b1, a1f);
    pool1<<<(BEEG * 25 * 247 + 255) / 256, 256, 0, stream>>>(a1f, a1p);

    // --- layers 2-4 fused conv+bn+elu+pool ---
    conv_pool<<<(BEEG * 50 * 79 + 255) / 256, 256, 0, stream>>>(a1p, w2f, b2f, a2, 25, 247, 50, 79);
    conv_pool<<<(BEEG * 100 * 23 + 255) / 256, 256, 0, stream>>>(a2, w3f, b3f, a3, 50, 79, 100, 23);
    conv_pool<<<(BEEG * 200 * 4 + 255) / 256, 256, 0, stream>>>(a3, w4f, b4f, a4, 100, 23, 200, 4);

    // --- fc + MoE heads ---
    fc_heads<<<BEEG, 128, 0, stream>>>(a4, fc_w, fc_b, cid, hW1, hb1, hW2, hb2, out);
}